// MSDeformAttnNoOutputProj_28467043238166
// MI455X (gfx1250) — compile-verified
//
#include <hip/hip_runtime.h>
#include <hip/hip_bf16.h>
#include <math.h>
#include <stdint.h>

typedef __attribute__((ext_vector_type(2))) float v2f;
typedef __attribute__((ext_vector_type(8))) float v8f;

#define NB      4        // batch
#define LQ      10000    // queries
#define CM      256      // d_model
#define NH      8        // heads
#define NL      4        // levels
#define NP      2        // points
#define HD      32       // head dim
#define S_TOT   19560    // sum of H*W over levels

// LDS A-panel stride: 260 % 64 == 4 -> lanes 0-15 hit banks kk+4m (distinct),
// lanes 16-31 hit banks kk+2+4m (distinct, disjoint) => conflict-free b64 reads.
// 260 is a multiple of 4 floats, so 16B async chunks never straddle the pad.
#define AS_LD   260

// Async-copy one 16B chunk from global memory directly into LDS (no VGPR staging).
__device__ __forceinline__ void async_copy16(const float* gsrc, float* ldst) {
    unsigned int       ldsOff = (unsigned int)(uintptr_t)ldst;      // LDS aperture: low 32 bits = LDS byte offset
    unsigned long long gaddr  = (unsigned long long)(uintptr_t)gsrc;
    asm volatile("global_load_async_to_lds_b128 %0, %1, off"
                 :: "v"(ldsOff), "v"(gaddr) : "memory");
}
__device__ __forceinline__ void async_wait0() {
    asm volatile("s_wait_asynccnt 0" ::: "memory");
}

// ---------------------------------------------------------------------------
// Kernel 1: value = input_flatten @ W_val + b_val, scattered to (N, NH, S, HD)
// fp32 WMMA 16x16x4. Block = 256 thr = 8 waves; each wave owns 2 col-tiles,
// K-loop outer so one A fragment (1 ds_load_b64) feeds two WMMAs.
// ---------------------------------------------------------------------------
__global__ __launch_bounds__(256) void value_gemm_kernel(
    const float* __restrict__ A,     // (NB*S_TOT, 256) input_flatten
    const float* __restrict__ Wv,    // (256, 256)
    const float* __restrict__ bv,    // (256)
    float* __restrict__ value)       // (NB, NH, S_TOT, HD)
{
    __shared__ float As[16 * AS_LD];
    const int tid     = threadIdx.x;
    const int rowBase = blockIdx.x * 16;

    // Async-stage 16x256 fp32 A panel into LDS: 1024 16B chunks, 4 per thread.
    #pragma unroll
    for (int c = tid; c < 1024; c += 256) {
        const int r  = c >> 6;           // 64 float4 chunks per row
        const int c4 = (c & 63) << 2;    // float column
        async_copy16(A + (size_t)(rowBase + r) * CM + c4, &As[r * AS_LD + c4]);
    }
    async_wait0();
    __syncthreads();

    const int wave = tid >> 5;
    const int lane = tid & 31;
    const int half = lane >> 4;          // 0: K+0/K+1,  1: K+2/K+3 (A & B layout)
    const int mrow = lane & 15;          // A row (M)
    const int col0 = wave * 32 + (lane & 15);
    const int col1 = col0 + 16;
    const float* __restrict__ as_row = &As[mrow * AS_LD];

    v8f acc0 = {}, acc1 = {};
    for (int k = 0; k < CM; k += 4) {
        const int kk = k + half * 2;
        v2f a, b0, b1;
        a.x  = as_row[kk];
        a.y  = as_row[kk + 1];
        b0.x = Wv[(size_t)kk * CM + col0];
        b0.y = Wv[(size_t)(kk + 1) * CM + col0];
        b1.x = Wv[(size_t)kk * CM + col1];
        b1.y = Wv[(size_t)(kk + 1) * CM + col1];
        acc0 = __builtin_amdgcn_wmma_f32_16x16x4_f32(
            false, a, false, b0, (short)0, acc0, false, false);
        acc1 = __builtin_amdgcn_wmma_f32_16x16x4_f32(
            false, a, false, b1, (short)0, acc1, false, false);
    }

    // Epilogue: scatter to (N, NH, S, HD). A 16-row tile crosses the batch
    // boundary at most once -> scalar n0 + one compare/select per row.
    const int n0  = rowBase / S_TOT;
    const int rem = rowBase - n0 * S_TOT;
    #pragma unroll
    for (int t = 0; t < 2; ++t) {
        const int col = t ? col1 : col0;
        const float bc = bv[col];
        const int h = col >> 5, d = col & 31;
        #pragma unroll
        for (int v = 0; v < 8; ++v) {
            int s = rem + v + half * 8;
            int n = n0;
            if (s >= S_TOT) { n++; s -= S_TOT; }
            const float cv = (t ? acc1[v] : acc0[v]) + bc;
            value[(((size_t)n * NH + h) * S_TOT + s) * HD + d] = cv;
        }
    }
}

// ---------------------------------------------------------------------------
// Kernel 2: [off | attn] = query @ [W_off | W_attn] + bias.  OUTC = 192.
// Block = 192 thr = 6 waves; waves 0-3 -> off cols, waves 4-5 -> attn cols.
// ---------------------------------------------------------------------------
__global__ __launch_bounds__(192) void offattn_gemm_kernel(
    const float* __restrict__ Q,      // (NB*LQ, 256)
    const float* __restrict__ Woff,   // (256, 128)
    const float* __restrict__ boff,   // (128)
    const float* __restrict__ Watt,   // (256, 64)
    const float* __restrict__ batt,   // (64)
    float* __restrict__ off,          // (NB*LQ, 128)
    float* __restrict__ attn)         // (NB*LQ, 64)
{
    __shared__ float As[16 * AS_LD];
    const int tid     = threadIdx.x;
    const int rowBase = blockIdx.x * 16;

    for (int c = tid; c < 1024; c += 192) {
        const int r  = c >> 6;
        const int c4 = (c & 63) << 2;
        async_copy16(Q + (size_t)(rowBase + r) * CM + c4, &As[r * AS_LD + c4]);
    }
    async_wait0();
    __syncthreads();

    const int wave = tid >> 5;
    const int lane = tid & 31;
    const int half = lane >> 4;
    const int mrow = lane & 15;

    const bool isOff = (wave < 4);
    const float* __restrict__ Wt = isOff ? Woff : Watt;
    const int ldw  = isOff ? 128 : 64;
    const int col0 = (isOff ? wave : wave - 4) * 32 + (lane & 15);
    const int col1 = col0 + 16;
    const float* __restrict__ as_row = &As[mrow * AS_LD];

    v8f acc0 = {}, acc1 = {};
    for (int k = 0; k < CM; k += 4) {
        const int kk = k + half * 2;
        v2f a, b0, b1;
        a.x  = as_row[kk];
        a.y  = as_row[kk + 1];
        b0.x = Wt[(size_t)kk * ldw + col0];
        b0.y = Wt[(size_t)(kk + 1) * ldw + col0];
        b1.x = Wt[(size_t)kk * ldw + col1];
        b1.y = Wt[(size_t)(kk + 1) * ldw + col1];
        acc0 = __builtin_amdgcn_wmma_f32_16x16x4_f32(
            false, a, false, b0, (short)0, acc0, false, false);
        acc1 = __builtin_amdgcn_wmma_f32_16x16x4_f32(
            false, a, false, b1, (short)0, acc1, false, false);
    }

    const float* __restrict__ bp = isOff ? boff : batt;
    float* __restrict__ outp     = isOff ? off : attn;
    #pragma unroll
    for (int t = 0; t < 2; ++t) {
        const int col = t ? col1 : col0;
        const float bc = bp[col];
        #pragma unroll
        for (int v = 0; v < 8; ++v) {
            const int row = rowBase + v + half * 8;
            outp[(size_t)row * ldw + col] = (t ? acc1[v] : acc0[v]) + bc;
        }
    }
}

// ---------------------------------------------------------------------------
// Kernel 3: softmax over 8 (l,p) + bilinear gather-accumulate.
// One wave32 per (n, q, h); lane = head channel. Each corner gather is one
// coalesced 128B line from the L2-resident (N,NH,S,HD) value tensor.
// ---------------------------------------------------------------------------
__global__ __launch_bounds__(256) void sample_kernel(
    const float* __restrict__ refp,   // (NB, LQ, 2, 2)
    const float* __restrict__ off,    // (NB*LQ, 128)
    const float* __restrict__ attnl,  // (NB*LQ, 64)
    const float* __restrict__ value,  // (NB, NH, S_TOT, HD)
    float* __restrict__ out)          // (NB, LQ, 256)
{
    const int wid  = (blockIdx.x * blockDim.x + threadIdx.x) >> 5;
    const int lane = threadIdx.x & 31;
    if (wid >= NB * LQ * NH) return;
    const int h  = wid & 7;
    const int nq = wid >> 3;               // n*LQ + q
    const int n  = nq / LQ;

    // softmax over the 8 logits for this (n,q,h) -- broadcast loads
    const float* al = attnl + (size_t)nq * 64 + h * 8;
    float e[8];
    float m = al[0];
    #pragma unroll
    for (int j = 1; j < 8; ++j) m = fmaxf(m, al[j]);
    float sum = 0.f;
    #pragma unroll
    for (int j = 0; j < 8; ++j) { e[j] = __expf(al[j] - m); sum += e[j]; }
    const float inv = 1.f / sum;

    const float rx[2] = { refp[(size_t)nq * 4 + 0], refp[(size_t)nq * 4 + 2] };
    const float ry[2] = { refp[(size_t)nq * 4 + 1], refp[(size_t)nq * 4 + 3] };

    const int Hs[NL] = { 92, 46, 23, 12 };
    const int Ws[NL] = { 160, 80, 40, 20 };
    const int St[NL] = { 0, 14720, 18400, 19320 };

    const float* __restrict__ vbase = value + ((size_t)n * NH + h) * S_TOT * HD;
    const float* __restrict__ offq  = off + (size_t)nq * 128 + h * 16;

    float acc = 0.f;
    #pragma unroll
    for (int l = 0; l < NL; ++l) {
        const int   Wl = Ws[l], Hl = Hs[l];
        const float Wf = (float)Wl, Hf = (float)Hl;
        const float* __restrict__ vl = vbase + (size_t)St[l] * HD;
        #pragma unroll
        for (int p = 0; p < NP; ++p) {
            const float ox = offq[l * 4 + p * 2 + 0];
            const float oy = offq[l * 4 + p * 2 + 1];
            const float x  = (rx[p] + ox / Wf) * Wf - 0.5f;
            const float y  = (ry[p] + oy / Hf) * Hf - 0.5f;
            const float x0f = floorf(x), y0f = floorf(y);
            const float fx = x - x0f, fy = y - y0f;
            const int x0 = (int)x0f, y0 = (int)y0f;
            const float aw  = e[l * 2 + p] * inv;
            const float w00 = (1.f - fx) * (1.f - fy) * aw;
            const float w10 = fx * (1.f - fy) * aw;
            const float w01 = (1.f - fx) * fy * aw;
            const float w11 = fx * fy * aw;
            // zero-padded bilinear: skip OOB corners (uniform branches per wave)
            if (x0 >= 0     && x0 < Wl     && y0 >= 0     && y0 < Hl)
                acc += w00 * vl[((size_t)y0 * Wl + x0) * HD + lane];
            if (x0 + 1 >= 0 && x0 + 1 < Wl && y0 >= 0     && y0 < Hl)
                acc += w10 * vl[((size_t)y0 * Wl + x0 + 1) * HD + lane];
            if (x0 >= 0     && x0 < Wl     && y0 + 1 >= 0 && y0 + 1 < Hl)
                acc += w01 * vl[((size_t)(y0 + 1) * Wl + x0) * HD + lane];
            if (x0 + 1 >= 0 && x0 + 1 < Wl && y0 + 1 >= 0 && y0 + 1 < Hl)
                acc += w11 * vl[((size_t)(y0 + 1) * Wl + x0 + 1) * HD + lane];
        }
    }
    out[(size_t)nq * CM + h * HD + lane] = acc;
}

// ---------------------------------------------------------------------------
extern "C" void kernel_launch(void* const* d_in, const int* in_sizes, int n_in,
                              void* d_out, int out_size, void* d_ws, size_t ws_size,
                              hipStream_t stream) {
    const float* query  = (const float*)d_in[0];
    const float* refp   = (const float*)d_in[1];
    const float* inputf = (const float*)d_in[2];
    const float* W_off  = (const float*)d_in[3];
    const float* b_off  = (const float*)d_in[4];
    const float* W_attn = (const float*)d_in[5];
    const float* b_attn = (const float*)d_in[6];
    const float* W_val  = (const float*)d_in[7];
    const float* b_val  = (const float*)d_in[8];
    float* out = (float*)d_out;

    // Workspace layout (floats): value 20,029,440 | off 5,120,000 | attn 2,560,000
    float* value = (float*)d_ws;
    float* off   = value + (size_t)NB * NH * S_TOT * HD;
    float* attn  = off + (size_t)NB * LQ * (NH * NL * NP * 2);

    // value projection (78240x256x256 fp32 WMMA GEMM) -> (N,NH,S,HD)
    value_gemm_kernel<<<(NB * S_TOT) / 16, 256, 0, stream>>>(inputf, W_val, b_val, value);
    // sampling offsets + attention logits (40000x256x192 fp32 WMMA GEMM)
    offattn_gemm_kernel<<<(NB * LQ) / 16, 192, 0, stream>>>(
        query, W_off, b_off, W_attn, b_attn, off, attn);
    // softmax + deformable bilinear aggregation (one wave per (n,q,h))
    sample_kernel<<<(NB * LQ * NH) / 8, 256, 0, stream>>>(refp, off, attn, value, out);
}